// GCNModel_19275813224530
// MI455X (gfx1250) — compile-verified
//
#include <hip/hip_runtime.h>
#include <hip/hip_bf16.h>
#include <stdint.h>

// ---------------------------------------------------------------------------
// GCN (3x GCNConv) for MI455X / gfx1250, wave32.
//   N=100000 nodes, E=1250000 edges, dims 128 -> 64 -> 64 -> 16.
// Dense transforms use V_WMMA_F32_16X16X4_F32 (pure f32, matches reference
// precision). Edge aggregation uses float4 gathers + global_atomic_add_f32.
// W is repacked in LDS into WMMA fragment order so each B fragment is a
// single aligned ds_load_b64 (no cross-register packing movs).
// ---------------------------------------------------------------------------

#define GCN_N     100000
#define GCN_E     1250000
#define GCN_INF   128
#define GCN_HID   64
#define GCN_OUTF  16

typedef __attribute__((ext_vector_type(2))) float v2f;
typedef __attribute__((ext_vector_type(8))) float v8f;

// ---------------------------------------------------------------------------
// Degree / norm precompute
// ---------------------------------------------------------------------------
__global__ void k_deg_init(float* __restrict__ deg, int n) {
    int i = blockIdx.x * blockDim.x + threadIdx.x;
    if (i < n) deg[i] = 1.0f;                 // self-loop weight
}

__global__ void k_deg_edges(const long long* __restrict__ ei,
                            const float* __restrict__ ew,
                            float* __restrict__ deg, int ne) {
    int e = blockIdx.x * blockDim.x + threadIdx.x;
    if (e >= ne) return;
    int c = (int)ei[(size_t)ne + e];          // edge_index[1][e]
    atomicAdd(&deg[c], ew[e]);
}

__global__ void k_dinv(float* __restrict__ deg, int n) {
    int i = blockIdx.x * blockDim.x + threadIdx.x;
    if (i >= n) return;
    float d = deg[i];
    deg[i] = (d > 0.0f) ? rsqrtf(d) : 0.0f;   // in place: deg -> dinv
}

__global__ void k_norm(const long long* __restrict__ ei,
                       const float* __restrict__ ew,
                       const float* __restrict__ dinv,
                       float* __restrict__ norm, int ne) {
    int e = blockIdx.x * blockDim.x + threadIdx.x;
    if (e >= ne) return;
    int r = (int)ei[e];
    int c = (int)ei[(size_t)ne + e];
    norm[e] = dinv[r] * ew[e] * dinv[c];
}

// ---------------------------------------------------------------------------
// WMMA f32 GEMM: H[nrows x ODIM] = X[nrows x KDIM] @ W[KDIM x ODIM]
// Block = 256 threads = 8 waves; each wave owns one 16-row strip and
// ODIM/16 accumulator tiles. W staged in LDS in *fragment order*:
//   Wl[((k/4)*ODIM + n)*4 + (k%4)] = W[k][n]
// so lane (half,n) reads its B pair {W[k+2h][n], W[k+2h+1][n]} with one
// aligned ds_load_b64 at ((k/4)*ODIM + n)*16 + half*8 bytes.
// A-frag layout (16x4 f32): lane<16 -> K=0/1, lane>=16 -> K=2/3, M=lane&15.
// D layout (16x16 f32):     VGPR r -> M = r + 8*(lane>=16), N = lane&15.
// ---------------------------------------------------------------------------
template <int KDIM, int ODIM>
__global__ __launch_bounds__(256) void k_gemm_wmma(const float* __restrict__ X,
                                                   const float* __restrict__ W,
                                                   float* __restrict__ H,
                                                   int nrows) {
    constexpr int NT = ODIM / 16;
    __shared__ float Wl[KDIM * ODIM];

    // Cooperative fill with fragment repack. i steps by 4 within one row of W
    // (ODIM is a multiple of 4), so a float4 covers columns n0..n0+3 of row kk.
    for (int i = threadIdx.x * 4; i < KDIM * ODIM; i += 256 * 4) {
        float4 w4 = *reinterpret_cast<const float4*>(&W[i]);
        int kk = i / ODIM;
        int n0 = i - kk * ODIM;
        int kb = kk >> 2;
        int kr = kk & 3;
        float* dst = &Wl[(kb * ODIM + n0) * 4 + kr];
        dst[0]  = w4.x;
        dst[4]  = w4.y;
        dst[8]  = w4.z;
        dst[12] = w4.w;
    }
    __syncthreads();

    const int wave = threadIdx.x >> 5;
    const int lane = threadIdx.x & 31;
    const int half = lane >> 4;               // 0: lanes 0-15, 1: lanes 16-31
    const int m    = lane & 15;

    const int row0 = (blockIdx.x * 8 + wave) * 16;
    if (row0 >= nrows) return;

    v8f acc[NT] = {};
    const float* Xr = X + (size_t)(row0 + m) * KDIM;

    for (int k = 0; k < KDIM; k += 4) {
        v2f a = *reinterpret_cast<const v2f*>(&Xr[k + 2 * half]);   // b64 load
        const int kb = k >> 2;
#pragma unroll
        for (int t = 0; t < NT; ++t) {
            v2f b = *reinterpret_cast<const v2f*>(
                &Wl[(kb * ODIM + t * 16 + m) * 4 + 2 * half]);      // ds_load_b64
            acc[t] = __builtin_amdgcn_wmma_f32_16x16x4_f32(
                /*neg_a=*/false, a, /*neg_b=*/false, b,
                /*c_mod=*/(short)0, acc[t],
                /*reuse_a=*/false, /*reuse_b=*/false);
        }
    }

#pragma unroll
    for (int t = 0; t < NT; ++t) {
#pragma unroll
        for (int r = 0; r < 8; ++r) {
            H[(size_t)(row0 + 8 * half + r) * ODIM + t * 16 + m] = acc[t][r];
        }
    }
}

// ---------------------------------------------------------------------------
// agg[i][c] = h[i][c]*dinv[i]^2 + bias[c]   (self-loop term + bias; the
// remaining edge sum is linear so bias can be folded here)
// ---------------------------------------------------------------------------
template <int D>
__global__ void k_agg_init(const float* __restrict__ h,
                           const float* __restrict__ dinv,
                           const float* __restrict__ bias,
                           float* __restrict__ agg, int n) {
    constexpr int D4 = D / 4;
    int tid = blockIdx.x * blockDim.x + threadIdx.x;
    if (tid >= n * D4) return;
    int i = tid / D4;
    int q = tid - i * D4;
    float s = dinv[i];
    s = s * s;
    float4 hv = reinterpret_cast<const float4*>(h)[tid];
    float4 bv = reinterpret_cast<const float4*>(bias)[q];
    float4 o;
    o.x = hv.x * s + bv.x;
    o.y = hv.y * s + bv.y;
    o.z = hv.z * s + bv.z;
    o.w = hv.w * s + bv.w;
    reinterpret_cast<float4*>(agg)[tid] = o;
}

// ---------------------------------------------------------------------------
// Edge scatter: agg[col[e]] += h[row[e]] * norm[e]
// D/4 threads per edge, each handles a float4 slice -> a group of D/4
// consecutive lanes makes one contiguous 4*D-byte gather + atomic burst.
// ---------------------------------------------------------------------------
template <int D>
__global__ void k_edge_scatter(const long long* __restrict__ ei,
                               const float* __restrict__ norm,
                               const float* __restrict__ h,
                               float* __restrict__ agg, int ne) {
    constexpr int TPE = D / 4;
    int tid = blockIdx.x * blockDim.x + threadIdx.x;
    int e = tid / TPE;
    int q = tid - e * TPE;
    if (e >= ne) return;
    int r = (int)ei[e];
    int c = (int)ei[(size_t)ne + e];
    float s = norm[e];
    float4 hv = reinterpret_cast<const float4*>(h)[(size_t)r * TPE + q];
    float* base = agg + (size_t)c * D + q * 4;
    atomicAdd(base + 0, hv.x * s);
    atomicAdd(base + 1, hv.y * s);
    atomicAdd(base + 2, hv.z * s);
    atomicAdd(base + 3, hv.w * s);
}

__global__ void k_relu4(float* __restrict__ p, int n4) {
    int tid = blockIdx.x * blockDim.x + threadIdx.x;
    if (tid >= n4) return;
    float4 v = reinterpret_cast<float4*>(p)[tid];
    v.x = fmaxf(v.x, 0.0f);
    v.y = fmaxf(v.y, 0.0f);
    v.z = fmaxf(v.z, 0.0f);
    v.w = fmaxf(v.w, 0.0f);
    reinterpret_cast<float4*>(p)[tid] = v;
}

// ---------------------------------------------------------------------------
// Launch
// ---------------------------------------------------------------------------
extern "C" void kernel_launch(void* const* d_in, const int* in_sizes, int n_in,
                              void* d_out, int out_size, void* d_ws, size_t ws_size,
                              hipStream_t stream) {
    (void)in_sizes; (void)n_in; (void)out_size; (void)ws_size;

    const int N = GCN_N, E = GCN_E;

    const float*     x  = (const float*)d_in[0];
    const long long* ei = (const long long*)d_in[1];   // int64 edge_index [2,E]
    const float*     ew = (const float*)d_in[2];
    const float*     W1 = (const float*)d_in[3];
    const float*     b1 = (const float*)d_in[4];
    const float*     W2 = (const float*)d_in[5];
    const float*     b2 = (const float*)d_in[6];
    const float*     W3 = (const float*)d_in[7];
    const float*     b3 = (const float*)d_in[8];
    float*           out = (float*)d_out;

    // workspace layout (floats):  dinv[N] | norm[E] | A[N*64] | B[N*64]
    float* dinv = (float*)d_ws;
    float* nrm  = dinv + N;
    float* bufA = nrm + E;
    float* bufB = bufA + (size_t)N * GCN_HID;

    const int TB = 256;
    dim3 blk(TB);

    // ---- normalization ----
    k_deg_init <<<(N + TB - 1) / TB, blk, 0, stream>>>(dinv, N);
    k_deg_edges<<<(E + TB - 1) / TB, blk, 0, stream>>>(ei, ew, dinv, E);
    k_dinv     <<<(N + TB - 1) / TB, blk, 0, stream>>>(dinv, N);
    k_norm     <<<(E + TB - 1) / TB, blk, 0, stream>>>(ei, ew, dinv, nrm, E);

    const int gemm_grid = (N + 127) / 128;   // 8 waves * 16 rows per block

    // ---- layer 1: 128 -> 64 ----
    k_gemm_wmma<GCN_INF, GCN_HID><<<gemm_grid, blk, 0, stream>>>(x, W1, bufA, N);
    k_agg_init<GCN_HID><<<((N * (GCN_HID / 4)) + TB - 1) / TB, blk, 0, stream>>>(
        bufA, dinv, b1, bufB, N);
    k_edge_scatter<GCN_HID><<<((E * (GCN_HID / 4)) + TB - 1) / TB, blk, 0, stream>>>(
        ei, nrm, bufA, bufB, E);
    k_relu4<<<((N * GCN_HID / 4) + TB - 1) / TB, blk, 0, stream>>>(bufB, N * GCN_HID / 4);

    // ---- layer 2: 64 -> 64 ----
    k_gemm_wmma<GCN_HID, GCN_HID><<<gemm_grid, blk, 0, stream>>>(bufB, W2, bufA, N);
    k_agg_init<GCN_HID><<<((N * (GCN_HID / 4)) + TB - 1) / TB, blk, 0, stream>>>(
        bufA, dinv, b2, bufB, N);
    k_edge_scatter<GCN_HID><<<((E * (GCN_HID / 4)) + TB - 1) / TB, blk, 0, stream>>>(
        ei, nrm, bufA, bufB, E);
    k_relu4<<<((N * GCN_HID / 4) + TB - 1) / TB, blk, 0, stream>>>(bufB, N * GCN_HID / 4);

    // ---- layer 3: 64 -> 16 (output, no ReLU) ----
    k_gemm_wmma<GCN_HID, GCN_OUTF><<<gemm_grid, blk, 0, stream>>>(bufB, W3, bufA, N);
    k_agg_init<GCN_OUTF><<<((N * (GCN_OUTF / 4)) + TB - 1) / TB, blk, 0, stream>>>(
        bufA, dinv, b3, out, N);
    k_edge_scatter<GCN_OUTF><<<((E * (GCN_OUTF / 4)) + TB - 1) / TB, blk, 0, stream>>>(
        ei, nrm, bufA, out, E);
}